// ComposerLayer_52201032516319
// MI455X (gfx1250) — compile-verified
//
#include <hip/hip_runtime.h>
#include <math.h>

// Problem constants (reference: B,S,H,O = 16,2048,512,16)
#define B_ 16
#define S_ 2048
#define H_ 512
#define O_ 16
#define RSQRT_H 0.04419417382415922f  // 1/sqrt(512)

typedef float v2f __attribute__((ext_vector_type(2)));
typedef float v8f __attribute__((ext_vector_type(8)));

// D = A(16x4 f32) * B(4x16 f32) + C, full f32 precision via the matrix core.
__device__ __forceinline__ v8f wmma_f32(v2f a, v2f b, v8f c) {
    return __builtin_amdgcn_wmma_f32_16x16x4_f32(
        /*neg_a=*/false, a, /*neg_b=*/false, b,
        /*c_mod=*/(short)0, c, /*reuse_a=*/false, /*reuse_b=*/false);
}

// ---------------------------------------------------------------------------
// K1: operator_queries oq[o,h] = sum_s ops[s,o]*Wq[h,s] + bq[h]
//     8192 outputs, K=2048. Also writes the second tuple output of d_out.
// ---------------------------------------------------------------------------
__global__ void k_oq(const float* __restrict__ ops, const float* __restrict__ Wq,
                     const float* __restrict__ bq, float* __restrict__ oq_ws,
                     float* __restrict__ oq_out) {
    int idx = blockIdx.x * blockDim.x + threadIdx.x;   // 8192 threads
    int o = idx & 15, h = idx >> 4;
    const float* wrow = Wq + (size_t)h * S_;           // contiguous in s
    float acc = 0.f;
    for (int s = 0; s < S_; ++s) acc = fmaf(ops[s * O_ + o], wrow[s], acc);
    acc += bq[h];
    oq_ws[o * H_ + h]  = acc;
    oq_out[o * H_ + h] = acc;
}

// ---------------------------------------------------------------------------
// K2: folded key projection: Wko[h',o] = sum_h Wk[h,h']*oq[o,h]
//     blo[o] = sum_h bk[h]*oq[o,h]   (so logits = (x*Wko + blo)/sqrt(H))
// ---------------------------------------------------------------------------
__global__ void k_wko(const float* __restrict__ Wk, const float* __restrict__ bk,
                      const float* __restrict__ oq, float* __restrict__ wko,
                      float* __restrict__ blo) {
    int idx = blockIdx.x * blockDim.x + threadIdx.x;   // 8192 threads
    int o = idx & 15, hp = idx >> 4;
    const float* oqr = oq + (size_t)o * H_;
    float acc = 0.f;
    for (int h = 0; h < H_; ++h) acc = fmaf(Wk[h * H_ + hp], oqr[h], acc);
    wko[hp * O_ + o] = acc;                            // layout [h'][o]
    if (idx < O_) {
        const float* oq2 = oq + (size_t)idx * H_;
        float a2 = 0.f;
        for (int h = 0; h < H_; ++h) a2 = fmaf(bk[h], oq2[h], a2);
        blo[idx] = a2;
    }
}

// ---------------------------------------------------------------------------
// K3: logits[r, o] = (x[r,:] . Wko[:,o] + blo[o]) / sqrt(H), r = b*S+s.
//     WMMA: M=16 rows of x, N=16 operators, K=512 in 4-wide steps.
//     Wko (32KB) staged once per block in LDS. 8 waves/block, 256 blocks.
// ---------------------------------------------------------------------------
__global__ void k_logits(const float* __restrict__ x, const float* __restrict__ wko,
                         const float* __restrict__ blo, float* __restrict__ logits) {
    __shared__ float lwko[H_ * O_];   // 32 KB
    __shared__ float lblo[O_];
    for (int i = threadIdx.x; i < H_ * O_; i += blockDim.x) lwko[i] = wko[i];
    if (threadIdx.x < O_) lblo[threadIdx.x] = blo[threadIdx.x];
    __syncthreads();

    int wave = threadIdx.x >> 5, lane = threadIdx.x & 31;
    int m = lane & 15, half = lane >> 4;               // A: row m; halves split K
    int r0 = blockIdx.x * 128 + wave * 16;
    const float* arow = x + (size_t)(r0 + m) * H_;

    v8f c = {};
    for (int k0 = 0; k0 < H_; k0 += 4) {
        int kk = k0 + 2 * half;
        v2f a = *(const v2f*)(arow + kk);              // A(m, kk..kk+1)
        v2f bf;                                        // B(kk..kk+1, n=m)
        bf.x = lwko[kk * O_ + m];
        bf.y = lwko[(kk + 1) * O_ + m];
        c = wmma_f32(a, bf, c);
    }
    float bn = lblo[m];                                // n = lane&15
    for (int j = 0; j < 8; ++j) {
        int rm = j + 8 * half;                         // C/D: M = j + 8*half
        logits[(size_t)(r0 + rm) * O_ + m] = (c[j] + bn) * RSQRT_H;
    }
}

// ---------------------------------------------------------------------------
// K4: relevance softmax over s per (b,o); fused w[b,o,s]=rel*ops and
//     wsum[b,o]=sum_s w. One block per (b,o) column of 2048.
// ---------------------------------------------------------------------------
__global__ void k_rel(const float* __restrict__ logits, const float* __restrict__ ops,
                      float* __restrict__ w, float* __restrict__ wsum) {
    __shared__ float red[256];
    int b = blockIdx.x >> 4, o = blockIdx.x & 15;
    int t = threadIdx.x;
    const float* lcol = logits + (size_t)b * S_ * O_ + o;

    float mx = -3.4e38f;
    for (int s = t; s < S_; s += 256) mx = fmaxf(mx, lcol[s * O_]);
    red[t] = mx; __syncthreads();
    for (int st = 128; st > 0; st >>= 1) { if (t < st) red[t] = fmaxf(red[t], red[t + st]); __syncthreads(); }
    mx = red[0]; __syncthreads();

    float sm = 0.f;
    for (int s = t; s < S_; s += 256) sm += __expf(lcol[s * O_] - mx);
    red[t] = sm; __syncthreads();
    for (int st = 128; st > 0; st >>= 1) { if (t < st) red[t] += red[t + st]; __syncthreads(); }
    float inv = 1.0f / red[0]; __syncthreads();

    float* wrow = w + ((size_t)b * O_ + o) * S_;       // layout [b][o][s]
    float ws = 0.f;
    for (int s = t; s < S_; s += 256) {
        float val = __expf(lcol[s * O_] - mx) * inv * ops[s * O_ + o];
        wrow[s] = val;
        ws += val;
    }
    red[t] = ws; __syncthreads();
    for (int st = 128; st > 0; st >>= 1) { if (t < st) red[t] += red[t + st]; __syncthreads(); }
    if (t == 0) wsum[b * O_ + o] = red[0];
}

// ---------------------------------------------------------------------------
// K5: G[b,o,h'] = sum_s w[b,o,s]*x[b,s,h'].  Per b: 16 x 2048 @ 2048 x 512.
//     One wave per (b, 16-wide h' tile): 512 waves = 64 blocks x 8 waves.
// ---------------------------------------------------------------------------
__global__ void k_g(const float* __restrict__ w, const float* __restrict__ x,
                    float* __restrict__ G) {
    int wid = blockIdx.x * 8 + (threadIdx.x >> 5);
    int lane = threadIdx.x & 31;
    int b = wid >> 5, n0 = (wid & 31) * 16;
    int m = lane & 15, half = lane >> 4;
    const float* arow = w + ((size_t)b * O_ + m) * S_;
    const float* xb = x + (size_t)b * S_ * H_;

    v8f c = {};
    for (int k0 = 0; k0 < S_; k0 += 4) {
        int kk = k0 + 2 * half;
        v2f a = *(const v2f*)(arow + kk);              // A(o=m, s=kk..kk+1)
        v2f bf;                                        // B(s=kk..kk+1, h'=n0+m)
        bf.x = xb[(size_t)kk * H_ + n0 + m];
        bf.y = xb[(size_t)(kk + 1) * H_ + n0 + m];
        c = wmma_f32(a, bf, c);
    }
    for (int j = 0; j < 8; ++j) {
        int rm = j + 8 * half;
        G[((size_t)b * O_ + rm) * H_ + n0 + m] = c[j];
    }
}

// ---------------------------------------------------------------------------
// K6: oo[b,o,h] = sum_h' G[b,o,h']*Wv[h,h'] + bv[h]*wsum[b,o].
//     Per b: 16 x 512 @ 512 x 512 (B fragments are contiguous float2 of Wv rows).
// ---------------------------------------------------------------------------
__global__ void k_oo(const float* __restrict__ G, const float* __restrict__ Wv,
                     const float* __restrict__ bv, const float* __restrict__ wsum,
                     float* __restrict__ oo) {
    int wid = blockIdx.x * 8 + (threadIdx.x >> 5);
    int lane = threadIdx.x & 31;
    int b = wid >> 5, n0 = (wid & 31) * 16;
    int m = lane & 15, half = lane >> 4;
    const float* arow = G + ((size_t)b * O_ + m) * H_;

    v8f c = {};
    for (int k0 = 0; k0 < H_; k0 += 4) {
        int kk = k0 + 2 * half;
        v2f a = *(const v2f*)(arow + kk);                          // A(o=m, h'=kk..)
        v2f bf = *(const v2f*)(Wv + (size_t)(n0 + m) * H_ + kk);   // B(h'=kk.., h=n0+m)
        c = wmma_f32(a, bf, c);
    }
    for (int j = 0; j < 8; ++j) {
        int rm = j + 8 * half;
        oo[((size_t)b * O_ + rm) * H_ + n0 + m] = c[j] + bv[n0 + m] * wsum[b * O_ + rm];
    }
}

// ---------------------------------------------------------------------------
// K7: operator softmax over o (16-wide rows, in LDS) then
//     out[b,s,h] = sum_o ow[b,s,o]*oo[b,o,h].  K=16 -> 4 WMMA steps.
//     One block per (b, 16-row s tile); 8 waves cover H=512 (4 tiles each).
// ---------------------------------------------------------------------------
__global__ void k_out(const float* __restrict__ logits, const float* __restrict__ oo,
                      float* __restrict__ out) {
    __shared__ float lds_ow[O_][O_];                  // [s-row][o]
    int b = blockIdx.x >> 7;
    int s0 = (blockIdx.x & 127) * 16;
    int t = threadIdx.x;
    if (t < 16) {
        const float* lrow = logits + (size_t)(b * S_ + s0 + t) * O_;
        float mx = -3.4e38f;
        for (int o = 0; o < O_; ++o) mx = fmaxf(mx, lrow[o]);
        float e[O_]; float sm = 0.f;
        for (int o = 0; o < O_; ++o) { e[o] = __expf(lrow[o] - mx); sm += e[o]; }
        float inv = 1.0f / sm;
        for (int o = 0; o < O_; ++o) lds_ow[t][o] = e[o] * inv;
    }
    __syncthreads();

    int wave = t >> 5, lane = t & 31;
    int m = lane & 15, half = lane >> 4;
    const float* oob = oo + (size_t)b * O_ * H_;
    for (int sub = 0; sub < 4; ++sub) {
        int n0 = (wave * 4 + sub) * 16;
        v8f c = {};
        for (int k0 = 0; k0 < O_; k0 += 4) {
            int kk = k0 + 2 * half;
            v2f a;                                    // A(s-row=m, o=kk..kk+1)
            a.x = lds_ow[m][kk];
            a.y = lds_ow[m][kk + 1];
            v2f bf;                                   // B(o=kk.., h=n0+m)
            bf.x = oob[(size_t)kk * H_ + n0 + m];
            bf.y = oob[(size_t)(kk + 1) * H_ + n0 + m];
            c = wmma_f32(a, bf, c);
        }
        for (int j = 0; j < 8; ++j) {
            int rm = j + 8 * half;
            out[(size_t)(b * S_ + s0 + rm) * H_ + n0 + m] = c[j];
        }
    }
}

// ---------------------------------------------------------------------------
// Workspace layout (floats):
//   oq:8192 | wko:8192 | blo:16(+pad) | wsum:256 | logits:524288 |
//   w:524288 | G:131072 | oo:131072   -> ~5.2 MB total
// ---------------------------------------------------------------------------
extern "C" void kernel_launch(void* const* d_in, const int* in_sizes, int n_in,
                              void* d_out, int out_size, void* d_ws, size_t ws_size,
                              hipStream_t stream) {
    const float* x   = (const float*)d_in[0];
    const float* Wv  = (const float*)d_in[1];
    const float* bv  = (const float*)d_in[2];
    const float* Wk  = (const float*)d_in[3];
    const float* bk  = (const float*)d_in[4];
    const float* Wq  = (const float*)d_in[5];
    const float* bq  = (const float*)d_in[6];
    const float* ops = (const float*)d_in[7];
    float* out = (float*)d_out;
    float* ws  = (float*)d_ws;

    float* oq     = ws;
    float* wko    = ws + 8192;
    float* blo    = ws + 16384;
    float* wsum   = ws + 16640;
    float* logits = ws + 16896;
    float* w      = logits + (size_t)B_ * S_ * O_;
    float* G      = w + (size_t)B_ * S_ * O_;
    float* oo     = G + (size_t)B_ * O_ * H_;
    float* oq_out = out + (size_t)B_ * S_ * H_;   // tuple output #2

    k_oq    <<<32,   256, 0, stream>>>(ops, Wq, bq, oq, oq_out);
    k_wko   <<<32,   256, 0, stream>>>(Wk, bk, oq, wko, blo);
    k_logits<<<256,  256, 0, stream>>>(x, wko, blo, logits);
    k_rel   <<<256,  256, 0, stream>>>(logits, ops, w, wsum);
    k_g     <<<64,   256, 0, stream>>>(w, x, G);
    k_oo    <<<64,   256, 0, stream>>>(G, Wv, bv, wsum, oo);
    k_out   <<<2048, 256, 0, stream>>>(logits, oo, out);
}